// CriticNetwork_7516192768273
// MI455X (gfx1250) — compile-verified
//
#include <hip/hip_runtime.h>
#include <math.h>

// Sizes from the reference
#define BENV 4096
#define NAG  16
#define AACT 8
#define DIN  128
#define NH1  64
#define NDP  64

typedef __attribute__((ext_vector_type(2))) float v2f;
typedef __attribute__((ext_vector_type(4))) float f4;
typedef __attribute__((ext_vector_type(8))) float v8f;

// ---- d_ws layout (floats) ----------------------------------------------------
#define WS_MEAN  0                         // 4096*128 = 524288
#define WS_APA   (BENV * DIN)              // 65536
#define WS_APP   (WS_APA + BENV * NAG)     // 65536
#define WS_V     (WS_APP + BENV * NAG)     // 64
#define WS_FLOATS (WS_V + 64)
#define WS_BYTES  ((size_t)WS_FLOATS * 4)

// =============================================================================
// Kernel 1: bandwidth phase. ~590K threads so HBM (23.3 TB/s) is saturated by
// sheer memory-level parallelism — the 32MB obs stream dominates total traffic.
//  blocks [0,2048)   : mean[b][d] = (1/16) sum_n obs[b,n,d]
//  blocks [2048,2304): A_r = act_r . Wv[64:72],  P_r = pi_r . Wv[64:72]
//  block  2304       : v = Wfc @ (Wattn[:64] + Wattn[64:])
// =============================================================================
__global__ __launch_bounds__(256) void critic_prep(
    const float* __restrict__ obs,  const float* __restrict__ pol,
    const float* __restrict__ act,  const float* __restrict__ Wfc,
    const float* __restrict__ Wattn,const float* __restrict__ Wv,
    float* __restrict__ ws)
{
    const int bid = blockIdx.x;
    if (bid < 2048) {
        int tid = bid * 256 + threadIdx.x;          // 0..524287
        int b = tid >> 7, d = tid & 127;
        const float* base = obs + (size_t)b * (NAG * DIN) + d;
        float s = 0.f;
        #pragma unroll
        for (int n = 0; n < NAG; ++n) s += base[(size_t)n * DIN];
        ws[WS_MEAN + tid] = s * (1.0f / 16.0f);
    } else if (bid < 2304) {
        int r = (bid - 2048) * 256 + threadIdx.x;   // 0..65535
        float wv8[8];
        #pragma unroll
        for (int i = 0; i < 8; ++i) wv8[i] = Wv[64 + i];
        f4 aA = *(const f4*)(act + (size_t)r * AACT);
        f4 aB = *(const f4*)(act + (size_t)r * AACT + 4);
        f4 pA = *(const f4*)(pol + (size_t)r * AACT);
        f4 pB = *(const f4*)(pol + (size_t)r * AACT + 4);
        ws[WS_APA + r] = aA.x*wv8[0] + aA.y*wv8[1] + aA.z*wv8[2] + aA.w*wv8[3]
                       + aB.x*wv8[4] + aB.y*wv8[5] + aB.z*wv8[6] + aB.w*wv8[7];
        ws[WS_APP + r] = pA.x*wv8[0] + pA.y*wv8[1] + pA.z*wv8[2] + pA.w*wv8[3]
                       + pB.x*wv8[4] + pB.y*wv8[5] + pB.z*wv8[6] + pB.w*wv8[7];
    } else {
        int c = threadIdx.x;
        if (c < 64) {
            float s = 0.f;
            for (int k = 0; k < 64; ++k)
                s += Wfc[c * 64 + k] * (Wattn[k] + Wattn[64 + k]);
            ws[WS_V + c] = s;
        }
    }
}

// =============================================================================
// Kernel 2: WMMA phase. One wave per 16-env tile (256 blocks). Reads the
// L2-resident 2.5MB prep output, runs the exact-fp32 GEMM chain with
// V_WMMA_F32_16X16X4_F32, then streams the 8MB broadcast outputs (stores
// don't need latency hiding — STOREcnt allows 63 outstanding).
// =============================================================================
__global__ __launch_bounds__(32) void critic_core(
    const float* __restrict__ ws,   const float* __restrict__ W1,
    const float* __restrict__ b1,   const float* __restrict__ W2,
    const float* __restrict__ b2,   const float* __restrict__ Wv,
    const float* __restrict__ bvp,  float* __restrict__ out_x,
    float* __restrict__ out_w)
{
    __shared__ float buf[2112];     // mean tile stride 132; reused for h/p stride 68
    __shared__ float xv[256];
    __shared__ float wvL[16], cbL[16], szL[16];

    const int lane = threadIdx.x;
    const int hi   = lane >> 4;      // K-pair half select (ISA 7.12.2)
    const int nn   = lane & 15;      // M/N index within half
    const int e0   = blockIdx.x * 16;

    const float* meanb = ws + WS_MEAN;
    const float* APa   = ws + WS_APA;
    const float* APp   = ws + WS_APP;
    const float* vvec  = ws + WS_V;

    // stage mean tile: coalesced 512B rows -> LDS stride 132 (conflict-free frags)
    for (int e = 0; e < 16; ++e) {
        f4 t = *(const f4*)(meanb + (size_t)(e0 + e) * DIN + lane * 4);
        *(f4*)(buf + e * 132 + lane * 4) = t;
    }

    // ---- GEMM1: h = relu(mean @ W1 + b1)  (16x128 @ 128x64) --------------------
    v8f a0 = {}, a1 = {}, a2 = {}, a3 = {};
    for (int k0 = 0; k0 < DIN; k0 += 4) {
        const float* ap = buf + nn * 132 + k0 + 2 * hi;
        v2f av; av.x = ap[0]; av.y = ap[1];
        const float* wp = W1 + (size_t)(k0 + 2 * hi) * 64 + nn;
        v2f b0v; b0v.x = wp[0];  b0v.y = wp[64];
        v2f b1v; b1v.x = wp[16]; b1v.y = wp[80];
        v2f b2v; b2v.x = wp[32]; b2v.y = wp[96];
        v2f b3v; b3v.x = wp[48]; b3v.y = wp[112];
        a0 = __builtin_amdgcn_wmma_f32_16x16x4_f32(false, av, false, b0v, (short)0, a0, false, false);
        a1 = __builtin_amdgcn_wmma_f32_16x16x4_f32(false, av, false, b1v, (short)0, a1, false, false);
        a2 = __builtin_amdgcn_wmma_f32_16x16x4_f32(false, av, false, b2v, (short)0, a2, false, false);
        a3 = __builtin_amdgcn_wmma_f32_16x16x4_f32(false, av, false, b3v, (short)0, a3, false, false);
    }
    {
        const float c0 = b1[nn], c1 = b1[16 + nn], c2 = b1[32 + nn], c3 = b1[48 + nn];
        #pragma unroll
        for (int r = 0; r < 8; ++r) {
            const int m = r + 8 * hi;
            float h0 = a0[r] + c0, h1 = a1[r] + c1, h2 = a2[r] + c2, h3 = a3[r] + c3;
            buf[m * 68 + nn]      = h0 > 0.f ? h0 : 0.f;
            buf[m * 68 + 16 + nn] = h1 > 0.f ? h1 : 0.f;
            buf[m * 68 + 32 + nn] = h2 > 0.f ? h2 : 0.f;
            buf[m * 68 + 48 + nn] = h3 > 0.f ? h3 : 0.f;
        }
    }

    // ---- GEMM2: p = h @ W2 + b2  (16x64 @ 64x64) -------------------------------
    v8f p0 = {}, p1 = {}, p2 = {}, p3 = {};
    for (int k0 = 0; k0 < NH1; k0 += 4) {
        const float* ap = buf + nn * 68 + k0 + 2 * hi;
        v2f av; av.x = ap[0]; av.y = ap[1];
        const float* wp = W2 + (size_t)(k0 + 2 * hi) * 64 + nn;
        v2f b0v; b0v.x = wp[0];  b0v.y = wp[64];
        v2f b1v; b1v.x = wp[16]; b1v.y = wp[80];
        v2f b2v; b2v.x = wp[32]; b2v.y = wp[96];
        v2f b3v; b3v.x = wp[48]; b3v.y = wp[112];
        p0 = __builtin_amdgcn_wmma_f32_16x16x4_f32(false, av, false, b0v, (short)0, p0, false, false);
        p1 = __builtin_amdgcn_wmma_f32_16x16x4_f32(false, av, false, b1v, (short)0, p1, false, false);
        p2 = __builtin_amdgcn_wmma_f32_16x16x4_f32(false, av, false, b2v, (short)0, p2, false, false);
        p3 = __builtin_amdgcn_wmma_f32_16x16x4_f32(false, av, false, b3v, (short)0, p3, false, false);
    }
    {
        const float c0 = b2[nn], c1 = b2[16 + nn], c2 = b2[32 + nn], c3 = b2[48 + nn];
        #pragma unroll
        for (int r = 0; r < 8; ++r) {
            const int m = r + 8 * hi;
            buf[m * 68 + nn]      = p0[r] + c0;
            buf[m * 68 + 16 + nn] = p1[r] + c1;
            buf[m * 68 + 32 + nn] = p2[r] + c2;
            buf[m * 68 + 48 + nn] = p3[r] + c3;
        }
    }

    // ---- per-env scalars: w_b = sigmoid(leaky_relu(p.v)), c = p.Wv[:64]+bv -----
    {
        const int e = nn;  // both halves compute redundantly; half 0 stores
        float t = 0.f, c = 0.f;
        for (int i = 0; i < 64; ++i) {
            float pv = buf[e * 68 + i];
            t += pv * vvec[i];
            c += pv * Wv[i];
        }
        float el = t >= 0.f ? t : 0.01f * t;
        float wb = 1.0f / (1.0f + __expf(-el));
        if (hi == 0) { wvL[e] = wb; cbL[e] = c + bvp[0]; }
    }

    // ---- per-env sum_j Z_j (A/P dots precomputed in kernel 1, L2-resident) -----
    {
        const int e = nn;
        const float* pa = APa + (size_t)(e0 + e) * NAG;
        const float* pp = APp + (size_t)(e0 + e) * NAG;
        float sa = 0.f, sp = 0.f;
        #pragma unroll
        for (int q = 0; q < 4; ++q) {
            f4 va = *(const f4*)(pa + q * 4);
            f4 vp = *(const f4*)(pp + q * 4);
            sa += va.x + va.y + va.z + va.w;
            sp += vp.x + vp.y + vp.z + vp.w;
        }
        float wb = wvL[e];
        if (hi == 0) szL[e] = wb * sa + (1.f - wb) * sp;
    }

    // ---- x[env][j] = c + (SumZ - Z_j + P_j)/16 ---------------------------------
    for (int it = 0; it < 8; ++it) {
        int idx = it * 32 + lane;                 // env*16+node within tile
        size_t g = (size_t)e0 * NAG + idx;
        float Ai = APa[g], Pi = APp[g];
        int env = idx >> 4;
        float wb = wvL[env];
        float Zj = wb * Ai + (1.f - wb) * Pi;
        xv[idx] = cbL[env] + (szL[env] - Zj + Pi) * (1.0f / 16.0f);
    }

    // ---- stream outputs: x broadcast over dst; w broadcast everywhere ----------
    float* ox = out_x + (size_t)e0 * 256;
    float* ow = out_w + (size_t)e0 * 256;
    for (int it = 0; it < 32; ++it) {
        int f = (it * 32 + lane) * 4;             // j fastest, coalesced b128
        int env = f >> 8;
        f4 xval = *(const f4*)(xv + env * 16 + (f & 15));
        float wb = wvL[env];
        f4 wval = { wb, wb, wb, wb };
        *(f4*)(ox + f) = xval;
        *(f4*)(ow + f) = wval;
    }
}

// =============================================================================
// Fallback: fully-fused single kernel (used only if d_ws is too small).
// =============================================================================
#define WAVE_SMEM (2112 + 256 + 256 + 256 + 16 + 16 + 16 + 64)

__global__ __launch_bounds__(128) void critic_fused(
    const float* __restrict__ obs,  const float* __restrict__ pol,
    const float* __restrict__ act,  const float* __restrict__ W1,
    const float* __restrict__ b1,   const float* __restrict__ W2,
    const float* __restrict__ b2,   const float* __restrict__ Wfc,
    const float* __restrict__ Wattn,const float* __restrict__ Wv,
    const float* __restrict__ bvp,  float* __restrict__ out_x,
    float* __restrict__ out_w)
{
    __shared__ float smem[4 * WAVE_SMEM];
    const int wave = threadIdx.x >> 5;
    const int lane = threadIdx.x & 31;
    const int hi   = lane >> 4;
    const int nn   = lane & 15;

    float* buf = smem + wave * WAVE_SMEM;
    float* Aj  = buf + 2112;
    float* Pj  = Aj  + 256;
    float* xv  = Pj  + 256;
    float* wvL = xv  + 256;
    float* cbL = wvL + 16;
    float* szL = cbL + 16;
    float* vL  = szL + 16;

    const int tile = blockIdx.x * 4 + wave;
    const int e0   = tile * 16;

    for (int e = 0; e < 16; ++e) {
        f4 s = {0.f, 0.f, 0.f, 0.f};
        const float* base = obs + ((size_t)(e0 + e) * NAG) * DIN + lane * 4;
        #pragma unroll
        for (int k = 0; k < NAG; ++k) {
            f4 t = *(const f4*)(base + (size_t)k * DIN);
            s.x += t.x; s.y += t.y; s.z += t.z; s.w += t.w;
        }
        const float inv = 1.0f / 16.0f;
        f4 r = { s.x * inv, s.y * inv, s.z * inv, s.w * inv };
        *(f4*)(buf + e * 132 + lane * 4) = r;
    }
    for (int c = lane; c < 64; c += 32) {
        float s = 0.f;
        for (int k = 0; k < 64; ++k)
            s += Wfc[c * 64 + k] * (Wattn[k] + Wattn[64 + k]);
        vL[c] = s;
    }

    v8f a0 = {}, a1 = {}, a2 = {}, a3 = {};
    for (int k0 = 0; k0 < DIN; k0 += 4) {
        const float* ap = buf + nn * 132 + k0 + 2 * hi;
        v2f av; av.x = ap[0]; av.y = ap[1];
        const float* wp = W1 + (size_t)(k0 + 2 * hi) * 64 + nn;
        v2f b0v; b0v.x = wp[0];  b0v.y = wp[64];
        v2f b1v; b1v.x = wp[16]; b1v.y = wp[80];
        v2f b2v; b2v.x = wp[32]; b2v.y = wp[96];
        v2f b3v; b3v.x = wp[48]; b3v.y = wp[112];
        a0 = __builtin_amdgcn_wmma_f32_16x16x4_f32(false, av, false, b0v, (short)0, a0, false, false);
        a1 = __builtin_amdgcn_wmma_f32_16x16x4_f32(false, av, false, b1v, (short)0, a1, false, false);
        a2 = __builtin_amdgcn_wmma_f32_16x16x4_f32(false, av, false, b2v, (short)0, a2, false, false);
        a3 = __builtin_amdgcn_wmma_f32_16x16x4_f32(false, av, false, b3v, (short)0, a3, false, false);
    }
    {
        const float c0 = b1[nn], c1 = b1[16 + nn], c2 = b1[32 + nn], c3 = b1[48 + nn];
        #pragma unroll
        for (int r = 0; r < 8; ++r) {
            const int m = r + 8 * hi;
            float h0 = a0[r] + c0, h1 = a1[r] + c1, h2 = a2[r] + c2, h3 = a3[r] + c3;
            buf[m * 68 + nn]      = h0 > 0.f ? h0 : 0.f;
            buf[m * 68 + 16 + nn] = h1 > 0.f ? h1 : 0.f;
            buf[m * 68 + 32 + nn] = h2 > 0.f ? h2 : 0.f;
            buf[m * 68 + 48 + nn] = h3 > 0.f ? h3 : 0.f;
        }
    }

    v8f p0 = {}, p1 = {}, p2 = {}, p3 = {};
    for (int k0 = 0; k0 < NH1; k0 += 4) {
        const float* ap = buf + nn * 68 + k0 + 2 * hi;
        v2f av; av.x = ap[0]; av.y = ap[1];
        const float* wp = W2 + (size_t)(k0 + 2 * hi) * 64 + nn;
        v2f b0v; b0v.x = wp[0];  b0v.y = wp[64];
        v2f b1v; b1v.x = wp[16]; b1v.y = wp[80];
        v2f b2v; b2v.x = wp[32]; b2v.y = wp[96];
        v2f b3v; b3v.x = wp[48]; b3v.y = wp[112];
        p0 = __builtin_amdgcn_wmma_f32_16x16x4_f32(false, av, false, b0v, (short)0, p0, false, false);
        p1 = __builtin_amdgcn_wmma_f32_16x16x4_f32(false, av, false, b1v, (short)0, p1, false, false);
        p2 = __builtin_amdgcn_wmma_f32_16x16x4_f32(false, av, false, b2v, (short)0, p2, false, false);
        p3 = __builtin_amdgcn_wmma_f32_16x16x4_f32(false, av, false, b3v, (short)0, p3, false, false);
    }
    {
        const float c0 = b2[nn], c1 = b2[16 + nn], c2 = b2[32 + nn], c3 = b2[48 + nn];
        #pragma unroll
        for (int r = 0; r < 8; ++r) {
            const int m = r + 8 * hi;
            buf[m * 68 + nn]      = p0[r] + c0;
            buf[m * 68 + 16 + nn] = p1[r] + c1;
            buf[m * 68 + 32 + nn] = p2[r] + c2;
            buf[m * 68 + 48 + nn] = p3[r] + c3;
        }
    }

    {
        const int e = nn;
        float t = 0.f, c = 0.f;
        for (int i = 0; i < 64; ++i) {
            float pv = buf[e * 68 + i];
            t += pv * vL[i];
            c += pv * Wv[i];
        }
        float el = t >= 0.f ? t : 0.01f * t;
        float wb = 1.0f / (1.0f + __expf(-el));
        if (hi == 0) { wvL[e] = wb; cbL[e] = c + bvp[0]; }
    }

    float wv8[8];
    #pragma unroll
    for (int i = 0; i < 8; ++i) wv8[i] = Wv[64 + i];
    for (int it = 0; it < 8; ++it) {
        int idx = it * 32 + lane;
        size_t row = (size_t)e0 * NAG + idx;
        f4 aA = *(const f4*)(act + row * AACT);
        f4 aB = *(const f4*)(act + row * AACT + 4);
        f4 pA = *(const f4*)(pol + row * AACT);
        f4 pB = *(const f4*)(pol + row * AACT + 4);
        float da = aA.x*wv8[0] + aA.y*wv8[1] + aA.z*wv8[2] + aA.w*wv8[3]
                 + aB.x*wv8[4] + aB.y*wv8[5] + aB.z*wv8[6] + aB.w*wv8[7];
        float dp = pA.x*wv8[0] + pA.y*wv8[1] + pA.z*wv8[2] + pA.w*wv8[3]
                 + pB.x*wv8[4] + pB.y*wv8[5] + pB.z*wv8[6] + pB.w*wv8[7];
        Aj[idx] = da; Pj[idx] = dp;
    }

    {
        const int e = nn;
        float sa = 0.f, sp = 0.f;
        for (int j = 0; j < 16; ++j) { sa += Aj[e * 16 + j]; sp += Pj[e * 16 + j]; }
        float wb = wvL[e];
        if (hi == 0) szL[e] = wb * sa + (1.f - wb) * sp;
    }

    for (int it = 0; it < 8; ++it) {
        int idx = it * 32 + lane;
        int env = idx >> 4;
        float wb = wvL[env];
        float Pji = Pj[idx];
        float Zj = wb * Aj[idx] + (1.f - wb) * Pji;
        xv[idx] = cbL[env] + (szL[env] - Zj + Pji) * (1.0f / 16.0f);
    }

    float* ox = out_x + (size_t)e0 * 256;
    float* ow = out_w + (size_t)e0 * 256;
    for (int it = 0; it < 32; ++it) {
        int f = (it * 32 + lane) * 4;
        int env = f >> 8;
        f4 xval = *(const f4*)(xv + env * 16 + (f & 15));
        float wb = wvL[env];
        f4 wval = { wb, wb, wb, wb };
        *(f4*)(ox + f) = xval;
        *(f4*)(ow + f) = wval;
    }
}

extern "C" void kernel_launch(void* const* d_in, const int* in_sizes, int n_in,
                              void* d_out, int out_size, void* d_ws, size_t ws_size,
                              hipStream_t stream) {
    const float* obs   = (const float*)d_in[0];
    const float* pol   = (const float*)d_in[1];
    const float* act   = (const float*)d_in[2];
    const float* W1    = (const float*)d_in[3];
    const float* b1    = (const float*)d_in[4];
    const float* W2    = (const float*)d_in[5];
    const float* b2    = (const float*)d_in[6];
    const float* Wfc   = (const float*)d_in[7];
    const float* Wattn = (const float*)d_in[8];
    const float* Wv    = (const float*)d_in[9];
    const float* bv    = (const float*)d_in[10];
    float* out = (float*)d_out;
    float* out_w = out + (size_t)BENV * NAG * NAG;   // x first, then w

    if (ws_size >= WS_BYTES) {
        float* ws = (float*)d_ws;
        critic_prep<<<2305, 256, 0, stream>>>(obs, pol, act, Wfc, Wattn, Wv, ws);
        critic_core<<<256, 32, 0, stream>>>(ws, W1, b1, W2, b2, Wv, bv, out, out_w);
    } else {
        critic_fused<<<64, 128, 0, stream>>>(obs, pol, act, W1, b1, W2, b2,
                                             Wfc, Wattn, Wv, bv, out, out_w);
    }
}